// HypergraphRayleighQuotientLossDirect_60773787238818
// MI455X (gfx1250) — compile-verified
//
#include <hip/hip_runtime.h>
#include <hip/hip_bf16.h>

typedef __attribute__((ext_vector_type(2))) float v2f;
typedef __attribute__((ext_vector_type(8))) float v8f;

#define K_DIM 64
#define EPS 1e-8f

// ---------------------------------------------------------------------------
// K0: zero the workspace (must run every launch: graph replays reuse d_ws and
// all downstream stages accumulate via atomics).
// ---------------------------------------------------------------------------
__global__ void hrq_zero_kernel(float* __restrict__ p, long n) {
    long i = (long)blockIdx.x * blockDim.x + threadIdx.x;
    long stride = (long)gridDim.x * blockDim.x;
    for (; i < n; i += stride) p[i] = 0.0f;
}

// ---------------------------------------------------------------------------
// K1: degrees. Dv[n] += w[e], De[e] += 1 per pin. Scalar f32 atomics; both
// arrays (400KB + 200KB) are L2-resident.
// ---------------------------------------------------------------------------
__global__ void hrq_degrees_kernel(const int* __restrict__ node_idx,
                                   const int* __restrict__ edge_idx,
                                   const float* __restrict__ w,
                                   float* __restrict__ Dv,
                                   float* __restrict__ De,
                                   int nnz) {
    int i = blockIdx.x * blockDim.x + threadIdx.x;
    int stride = gridDim.x * blockDim.x;
    for (; i < nnz; i += stride) {
        int e = edge_idx[i];
        int n = node_idx[i];
        atomicAdd(&Dv[n], w[e]);
        atomicAdd(&De[e], 1.0f);
    }
}

// ---------------------------------------------------------------------------
// K1b: vw[e] = w[e] / De'[e] precomputed once (v_rcp_f32, not the 10-op IEEE
// divide chain), zero-padded to Epad so the WMMA edge loop needs no guards.
// ---------------------------------------------------------------------------
__global__ void hrq_edgeweight_kernel(const float* __restrict__ w,
                                      const float* __restrict__ De,
                                      float* __restrict__ vw,
                                      int E, int Epad) {
    int e = blockIdx.x * blockDim.x + threadIdx.x;
    if (e >= Epad) return;
    if (e < E) {
        float de = De[e];
        de = (de == 0.0f) ? 1.0f : de;
        vw[e] = w[e] * __builtin_amdgcn_rcpf(de);
    } else {
        vw[e] = 0.0f;
    }
}

// ---------------------------------------------------------------------------
// K2: scatter S[e,:] += Z[n,:] * rsqrt(Dv'[n]).  One wave32 per pin: each lane
// owns 2 columns (float2), giving one coalesced 256B row read of Z and 64
// contiguous global_atomic_add_f32 into one S row. S (12.8MB) stays in L2.
// ---------------------------------------------------------------------------
__global__ void hrq_scatter_kernel(const float* __restrict__ Z,
                                   const int* __restrict__ node_idx,
                                   const int* __restrict__ edge_idx,
                                   const float* __restrict__ Dv,
                                   float* __restrict__ S,
                                   int nnz) {
    int lane = threadIdx.x & 31;
    int waveInBlock = threadIdx.x >> 5;
    int wavesPerBlock = blockDim.x >> 5;
    long pin = (long)blockIdx.x * wavesPerBlock + waveInBlock;
    long pinStride = (long)gridDim.x * wavesPerBlock;
    for (; pin < nnz; pin += pinStride) {
        int n = node_idx[pin];  // wave-uniform load (broadcast)
        int e = edge_idx[pin];
        float dv = Dv[n];
        float rs = rsqrtf(dv == 0.0f ? 1.0f : dv);
        const float2* zrow = (const float2*)(Z + (long)n * K_DIM);
        float2 z = zrow[lane];
        float* srow = S + (long)e * K_DIM + 2 * lane;
        atomicAdd(srow,     z.x * rs);
        atomicAdd(srow + 1, z.y * rs);
    }
}

// ---------------------------------------------------------------------------
// K3: edge-side reduction  edgeterm[k] = sum_e vw[e] * S[e,k]^2
// V_WMMA_F32_16X16X4_F32 with A = 16x4 ones, B = 4 edges x 16 cols of
// weighted squares; C accumulates across edge chunks. Edges are padded to a
// multiple of 4 with zeroed S rows and vw entries, so the loop is branch-free
// and EXEC is all-ones throughout (WMMA requirement).
// B layout (32-bit, 4x16): VGPR0 = {lanes 0-15: K=0, lanes 16-31: K=2},
//                          VGPR1 = {lanes 0-15: K=1, lanes 16-31: K=3}.
// ---------------------------------------------------------------------------
__global__ void hrq_edge_reduce_wmma(const float* __restrict__ S,
                                     const float* __restrict__ vw,
                                     float* __restrict__ edgeterm,  // [64]
                                     int Epad) {
    int lane = threadIdx.x & 31;
    int waveId = blockIdx.x * (blockDim.x >> 5) + (threadIdx.x >> 5);
    int nWaves = gridDim.x * (blockDim.x >> 5);
    int ct = waveId & 3;                 // column tile (16 cols each)
    int chunk0 = waveId >> 2;
    int chunkStride = nWaves >> 2;
    int nChunks = Epad >> 2;

    int nb = lane & 15;                  // N index within tile
    int kbase = (lane < 16) ? 0 : 2;     // K rows this lane supplies
    int col = ct * 16 + nb;

    v2f a; a.x = 1.0f; a.y = 1.0f;       // ones A-matrix
    v8f c = {};
    for (int ch = chunk0; ch < nChunks; ch += chunkStride) {
        int e0 = ch * 4 + kbase;
        long base = (long)e0 * K_DIM + col;
        float s0 = S[base];
        float s1 = S[base + K_DIM];
        float v0 = vw[e0];
        float v1 = vw[e0 + 1];
        v2f b;
        b.x = v0 * s0 * s0;
        b.y = v1 * s1 * s1;
        c = __builtin_amdgcn_wmma_f32_16x16x4_f32(
                false, a, false, b, (short)0, c, false, false);
    }
    if (lane < 16) atomicAdd(&edgeterm[col], c[0]);  // row M=0 = column sums
}

// ---------------------------------------------------------------------------
// K4: node-side reductions (two WMMA accumulators share the ones-A operand):
//   den[k]    = sum_n Z[n,k]^2 * Dv'[n]           (Dv'=1 where Dv==0)
//   direct[k] = sum_n Z[n,k]^2 * 1[Dv[n] > 0]     ( == sum_e w_e SQ[e,k] )
// Main loop runs unguarded over N&~3 rows; <=3 tail rows are folded in by 64
// scalar threads of block 0 (atomics commute with the WMMA partial adds).
// ---------------------------------------------------------------------------
__global__ void hrq_node_reduce_wmma(const float* __restrict__ Z,
                                     const float* __restrict__ Dv,
                                     float* __restrict__ direct_,  // [64]
                                     float* __restrict__ den,      // [64]
                                     int N) {
    int lane = threadIdx.x & 31;
    int waveId = blockIdx.x * (blockDim.x >> 5) + (threadIdx.x >> 5);
    int nWaves = gridDim.x * (blockDim.x >> 5);
    int ct = waveId & 3;
    int chunk0 = waveId >> 2;
    int chunkStride = nWaves >> 2;
    int nChunks = N >> 2;                // full chunks only (unguarded)

    int nb = lane & 15;
    int kbase = (lane < 16) ? 0 : 2;
    int col = ct * 16 + nb;

    v2f a; a.x = 1.0f; a.y = 1.0f;
    v8f cden = {};
    v8f cdir = {};
    for (int ch = chunk0; ch < nChunks; ch += chunkStride) {
        int n0 = ch * 4 + kbase;
        long base = (long)n0 * K_DIM + col;
        float z0 = Z[base];
        float z1 = Z[base + K_DIM];
        float dv0 = Dv[n0];
        float dv1 = Dv[n0 + 1];
        float z20 = z0 * z0, z21 = z1 * z1;
        v2f bden, bdir;
        bden.x = z20 * ((dv0 == 0.0f) ? 1.0f : dv0);
        bden.y = z21 * ((dv1 == 0.0f) ? 1.0f : dv1);
        bdir.x = (dv0 > 0.0f) ? z20 : 0.0f;
        bdir.y = (dv1 > 0.0f) ? z21 : 0.0f;
        cden = __builtin_amdgcn_wmma_f32_16x16x4_f32(
                   false, a, false, bden, (short)0, cden, false, false);
        cdir = __builtin_amdgcn_wmma_f32_16x16x4_f32(
                   false, a, false, bdir, (short)0, cdir, false, false);
    }
    if (lane < 16) {
        atomicAdd(&den[col], cden[0]);
        atomicAdd(&direct_[col], cdir[0]);
    }

    // Tail rows (N % 4), handled scalar by 64 threads of block 0.
    if (blockIdx.x == 0 && threadIdx.x < K_DIM) {
        int k = threadIdx.x;
        for (int r = N & ~3; r < N; ++r) {
            float z = Z[(long)r * K_DIM + k];
            float z2 = z * z;
            float dv = Dv[r];
            atomicAdd(&den[k], z2 * ((dv == 0.0f) ? 1.0f : dv));
            if (dv > 0.0f) atomicAdd(&direct_[k], z2);
        }
    }
}

// ---------------------------------------------------------------------------
// K5: finalize  out = mean_k (direct[k] - edgeterm[k]) / (den[k] + eps)
// ---------------------------------------------------------------------------
__global__ void hrq_finalize_kernel(const float* __restrict__ direct_,
                                    const float* __restrict__ edgeterm,
                                    const float* __restrict__ den,
                                    float* __restrict__ out) {
    __shared__ float sh[K_DIM];
    int k = threadIdx.x;  // 64 threads
    sh[k] = (direct_[k] - edgeterm[k]) / (den[k] + EPS);
    __syncthreads();
    if (k == 0) {
        float s = 0.0f;
        for (int i = 0; i < K_DIM; ++i) s += sh[i];
        out[0] = s * (1.0f / K_DIM);
    }
}

// ---------------------------------------------------------------------------
extern "C" void kernel_launch(void* const* d_in, const int* in_sizes, int n_in,
                              void* d_out, int out_size, void* d_ws, size_t ws_size,
                              hipStream_t stream) {
    const float* Z    = (const float*)d_in[0];           // [N, 64]
    const int*   hidx = (const int*)d_in[1];             // [2, NNZ]
    const float* w    = (const float*)d_in[3];           // [E]

    const int  N    = in_sizes[0] / K_DIM;
    const long nnz  = (long)in_sizes[1] / 2;
    const int  E    = in_sizes[3];
    const int  Epad = (E + 3) & ~3;

    const int* node_idx = hidx;
    const int* edge_idx = hidx + nnz;

    // Workspace (floats): Dv[N] | De[E] | vw[Epad] | S[Epad*64] | direct[64] | den[64] | eterm[64]
    float* Dv      = (float*)d_ws;
    float* De      = Dv + N;
    float* vw      = De + E;
    float* S       = vw + Epad;
    float* direct_ = S + (long)Epad * K_DIM;
    float* den     = direct_ + K_DIM;
    float* eterm   = den + K_DIM;

    const long zeroN = (long)N + E + Epad + (long)Epad * K_DIM + 3 * K_DIM;
    hrq_zero_kernel<<<1024, 256, 0, stream>>>((float*)d_ws, zeroN);

    hrq_degrees_kernel<<<2048, 256, 0, stream>>>(node_idx, edge_idx, w, Dv, De, (int)nnz);

    hrq_edgeweight_kernel<<<(Epad + 255) / 256, 256, 0, stream>>>(w, De, vw, E, Epad);

    const int pinsPerBlock = 8;  // 256 threads = 8 wave32, one pin per wave
    int scatterBlocks = (int)((nnz + pinsPerBlock - 1) / pinsPerBlock);
    hrq_scatter_kernel<<<scatterBlocks, 256, 0, stream>>>(Z, node_idx, edge_idx, Dv, S, (int)nnz);

    hrq_edge_reduce_wmma<<<512, 256, 0, stream>>>(S, vw, eterm, Epad);
    hrq_node_reduce_wmma<<<512, 256, 0, stream>>>(Z, Dv, direct_, den, N);

    hrq_finalize_kernel<<<1, K_DIM, 0, stream>>>(direct_, eterm, den, (float*)d_out);
}